// Attention_dot_42073499632270
// MI455X (gfx1250) — compile-verified
//
#include <hip/hip_runtime.h>
#include <hip/hip_bf16.h>

// Sliding-window (W=11) rank-128 attention over T=100000 fp32 rows.
// MI455X / gfx1250, wave32:
//  - One wave per 16-timestep tile; 8 waves (256 thr) per block share an LDS
//    tile of 138 rows (128 + 10 halo) -> HBM reads each row ~1.08x.
//  - Phase 0 uses GLOBAL_LOAD_ASYNC_TO_LDS_B128 (ASYNCcnt path, no VGPR
//    staging); OOB halo rows are zero-filled with ds_store.
//  - Banded Gram matrix via chained V_WMMA_F32_16X16X4_F32 (fp32-exact vs
//    reference): two 16x16 f32 accumulators over K=128 (64 WMMAs/tile).
//  - Softmax over the 11-logit window with v_exp_f32.
//  - Weighted neighbor sum with coalesced float4 LDS loads and NT global
//    stores (output is write-once).

typedef __attribute__((ext_vector_type(2))) float v2f;
typedef __attribute__((ext_vector_type(4))) float v4f;
typedef __attribute__((ext_vector_type(8))) float v8f;

#define T_DIM    100000
#define RANK     128
#define WIN      11
#define PADR     5            // (WIN-1)/2
#define TILE     16           // timesteps per wave
#define WAVES    8            // waves per block
#define NTHREADS (WAVES * 32)
#define ROWS_BLK (WAVES * TILE)          // 128 rows of output per block
#define LDS_ROWS (ROWS_BLK + WIN - 1)    // 138 rows incl. halo
#define XPITCH   (RANK + 4)              // pad: column reads hit 32 banks
#define GSTRIDE  33                      // 32 Gram rows + 1 pad

__global__ __launch_bounds__(NTHREADS)
void slidewin_attn_wmma(const float* __restrict__ in, float* __restrict__ out)
{
    __shared__ float Xs[LDS_ROWS][XPITCH];          // input tile (halo'd)
    __shared__ float Gs[WAVES][TILE][GSTRIDE];      // Gram scratch  [i][m]
    __shared__ float Ps[WAVES][TILE][WIN + 1];      // softmax probs [i][j]

    const int tid  = threadIdx.x;
    const int lane = tid & 31;
    const int w    = tid >> 5;
    const long long blockRow0 = (long long)blockIdx.x * ROWS_BLK;

    // ---------------- Phase 0: async global -> LDS (zero-padded halo) -----
    for (int idx = tid; idx < LDS_ROWS * (RANK / 4); idx += NTHREADS) {
        const int r  = idx >> 5;         // RANK/4 == 32 float4 per row
        const int c4 = idx & 31;
        const long long g = blockRow0 - PADR + r;
        if (g >= 0 && g < T_DIM) {
            const unsigned lds_off = (unsigned)(size_t)&Xs[r][c4 * 4];
            const float*   src     = in + g * RANK + c4 * 4;
            // mem -> LDS directly, tracked by ASYNCcnt (no VGPR staging)
            asm volatile("global_load_async_to_lds_b128 %0, %1, off"
                         :: "v"(lds_off), "v"(src) : "memory");
        } else {
            const v4f z = {0.f, 0.f, 0.f, 0.f};
            *(v4f*)(&Xs[r][c4 * 4]) = z;
        }
    }
    asm volatile("s_wait_asynccnt 0x0" ::: "memory");
    __syncthreads();

    // ---------------- Phase 1: banded Gram via V_WMMA_F32_16X16X4_F32 -----
    // G[m][i] = dot(X[t0-5+m], X[t0+i]);  m=0..25 used (26..31 clamped junk)
    const int base = w * TILE;                       // wave's first LDS row
    const long long t0 = blockRow0 + base;           // wave's first timestep
    const int n    = lane & 15;                      // A-row / B-col index
    const int koff = (lane >> 4) << 1;               // K sub-offset: 0 or 2
    const int rB   = base + PADR + n;                // B: X[t0+n]
    const int rA0  = base + n;                       // A half 0: m = n
    const int rA1  = base + TILE + (n < WIN - 1 ? n : WIN - 2); // clamp m>25

    v8f G0 = {};   // rows m = 0..15
    v8f G1 = {};   // rows m = 16..31 (only 16..25 consumed)
    #pragma unroll
    for (int kk = 0; kk < RANK / 4; ++kk) {
        const int k = (kk << 2) + koff;
        const v2f b  = *(const v2f*)(&Xs[rB ][k]);
        const v2f a0 = *(const v2f*)(&Xs[rA0][k]);
        const v2f a1 = *(const v2f*)(&Xs[rA1][k]);
        G0 = __builtin_amdgcn_wmma_f32_16x16x4_f32(false, a0, false, b,
                                                   (short)0, G0, false, false);
        G1 = __builtin_amdgcn_wmma_f32_16x16x4_f32(false, a1, false, b,
                                                   (short)0, G1, false, false);
    }

    // D layout: VGPR vv, lanes 0-15 -> (M=vv, N=lane); 16-31 -> (M=vv+8).
    const int hi8 = (lane >> 4) << 3;
    #pragma unroll
    for (int vv = 0; vv < 8; ++vv) {
        Gs[w][n][vv + hi8]        = G0[vv];
        Gs[w][n][TILE + vv + hi8] = G1[vv];
    }
    __syncthreads();

    // ---------------- Softmax over the 11-wide window (lanes 0..15) -------
    if (lane < TILE) {
        const int i = lane;
        float e[WIN];
        float mx = -3.4e38f;
        #pragma unroll
        for (int j = 0; j < WIN; ++j) {
            e[j] = Gs[w][i][i + j];          // cos[t0+i][j] = G[i+j][i]
            mx = fmaxf(mx, e[j]);
        }
        float s = 0.f;
        #pragma unroll
        for (int j = 0; j < WIN; ++j) { e[j] = __expf(e[j] - mx); s += e[j]; }
        const float inv = 1.f / s;
        #pragma unroll
        for (int j = 0; j < WIN; ++j) Ps[w][i][j] = e[j] * inv;
    }
    __syncthreads();

    // ---------------- Phase 2: weighted neighbor sum, coalesced stores ----
    // lane owns columns [lane*4, lane*4+4); result[t0+i] = sum_j p[j]*X[t0+i+j-5]
    if (t0 < T_DIM) {
        for (int i = 0; i < TILE; ++i) {
            v4f acc = {0.f, 0.f, 0.f, 0.f};
            #pragma unroll
            for (int j = 0; j < WIN; ++j) {
                const float p = Ps[w][i][j];
                const v4f  x = *(const v4f*)(&Xs[base + i + j][lane * 4]);
                acc = acc + x * p;
            }
            __builtin_nontemporal_store(
                acc, (v4f*)(out + (t0 + i) * RANK + (long long)lane * 4));
        }
    }
}

extern "C" void kernel_launch(void* const* d_in, const int* in_sizes, int n_in,
                              void* d_out, int out_size, void* d_ws, size_t ws_size,
                              hipStream_t stream)
{
    (void)in_sizes; (void)n_in; (void)d_ws; (void)ws_size; (void)out_size;
    const float* in  = (const float*)d_in[0];
    float*       out = (float*)d_out;

    const int nTiles  = (T_DIM + TILE - 1) / TILE;          // 6250
    const int nBlocks = (nTiles + WAVES - 1) / WAVES;       // 782
    slidewin_attn_wmma<<<nBlocks, NTHREADS, 0, stream>>>(in, out);
}